// ClusterTripletLoss_25228637896963
// MI455X (gfx1250) — compile-verified
//
#include <hip/hip_runtime.h>
#include <hip/hip_bf16.h>

// Problem constants (fixed by the reference).
#define B_ROWS 4096
#define K_CENT 1024
#define DIM    128
#define ROWS   16            // sample rows per block
#define CHUNK  64            // centroid rows staged in LDS per buffer (32 KB)
#define NCHUNK (K_CENT / CHUNK)
#define EPS    1e-6f
#define MARGIN 1.0f

typedef int v4i __attribute__((ext_vector_type(4)));
typedef __attribute__((address_space(1))) v4i* gv4i_ptr;   // global int4*
typedef __attribute__((address_space(3))) v4i* lv4i_ptr;   // LDS int4*

// gfx1250 async copy: global -> LDS, 16 bytes per lane, tracked by ASYNCcnt.
__device__ __forceinline__ void async_copy_b128(void* lds_dst, const void* gsrc) {
#if __has_builtin(__builtin_amdgcn_global_load_async_to_lds_b128)
  __builtin_amdgcn_global_load_async_to_lds_b128(
      (gv4i_ptr)(unsigned long long)(size_t)gsrc,   // flat==global bit pattern
      (lv4i_ptr)(unsigned)(size_t)lds_dst,          // low 32 bits = LDS address
      0, 0);
#else
  unsigned lds_off = (unsigned)(size_t)lds_dst;
  asm volatile("global_load_async_to_lds_b128 %0, %1, off"
               :: "v"(lds_off), "v"(gsrc) : "memory");
#endif
}

__device__ __forceinline__ void wait_async0() {
#if __has_builtin(__builtin_amdgcn_s_wait_asynccnt)
  __builtin_amdgcn_s_wait_asynccnt(0);
#else
  asm volatile("s_wait_asynccnt 0" ::: "memory");
#endif
}

__global__ __launch_bounds__(256)
void ctl_main_kernel(const float* __restrict__ x,
                     const float* __restrict__ cent,
                     float* __restrict__ block_loss) {
  __shared__ float sX[ROWS * DIM];         //  8 KB : input tile
  __shared__ float sC[2][CHUNK * DIM];     // 64 KB : double-buffered centroid chunks
  __shared__ int   sIMin[ROWS * DIM];      //  8 KB : per-(row,dim) argmin index
  __shared__ int   sIMax[ROWS * DIM];      //  8 KB : per-(row,dim) argmax index
  __shared__ int   sModeP[ROWS];
  __shared__ int   sModeN[ROWS];
  __shared__ float sLoss[ROWS];

  const int t       = threadIdx.x;        // 256 threads = 8 wave32
  const int d       = t & (DIM - 1);      // dimension this thread owns
  const int half    = t >> 7;             // 0..1 -> interleaved row set
  const int rowBase = blockIdx.x * ROWS;

  // ---- prologue: async-stage X tile (512 float4) and centroid chunk 0 ----
  for (int i = 0; i < 2; ++i) {
    const int q = t + i * 256;                         // float4 index
    async_copy_b128(&sX[q * 4], x + (size_t)rowBase * DIM + (size_t)q * 4);
  }
#pragma unroll
  for (int i = 0; i < 8; ++i) {
    const int q = t + i * 256;                         // 2048 float4 = 64x128 f32
    async_copy_b128(&sC[0][q * 4], cent + (size_t)q * 4);
  }
  wait_async0();
  __syncthreads();

  float xv[8];
#pragma unroll
  for (int r = 0; r < 8; ++r) xv[r] = sX[(half + 2 * r) * DIM + d];

  float bmin[8], bmax[8];
  int   imin[8], imax[8];
#pragma unroll
  for (int r = 0; r < 8; ++r) { bmin[r] = 3.4e38f; bmax[r] = -1.0f; imin[r] = 0; imax[r] = 0; }

  // ---- scan all K centroids; DMA of chunk ch+1 overlaps compute of ch ----
  for (int ch = 0; ch < NCHUNK; ++ch) {
    const int cur = ch & 1;
    if (ch + 1 < NCHUNK) {
      // Buffer cur^1 was last read in iteration ch-1, which ended with a
      // barrier, so every wave is done with it: safe to overwrite now.
#pragma unroll
      for (int i = 0; i < 8; ++i) {
        const int q = t + i * 256;
        async_copy_b128(&sC[cur ^ 1][q * 4],
                        cent + (size_t)(ch + 1) * CHUNK * DIM + (size_t)q * 4);
      }
    }

#pragma unroll 4
    for (int k = 0; k < CHUNK; ++k) {
      const float c  = sC[cur][k * DIM + d];   // one ds read feeds 8 rows
      const int   kg = ch * CHUNK + k;
#pragma unroll
      for (int r = 0; r < 8; ++r) {
        const float df = fabsf(xv[r] - c);          // same ordering as (x-c)^2
        const bool  lt = df < bmin[r];              // strict: first-min like argmin
        bmin[r] = lt ? df : bmin[r];
        imin[r] = lt ? kg : imin[r];
        const bool  gt = df > bmax[r];              // strict: first-max like argmax
        bmax[r] = gt ? df : bmax[r];
        imax[r] = gt ? kg : imax[r];
      }
    }

    if (ch + 1 < NCHUNK) {
      wait_async0();        // next chunk landed (ran concurrently with compute)
      __syncthreads();      // all waves: next chunk visible, this buffer free
    }
  }

#pragma unroll
  for (int r = 0; r < 8; ++r) {
    const int row = half + 2 * r;
    sIMin[row * DIM + d] = imin[r];
    sIMax[row * DIM + d] = imax[r];
  }
  __syncthreads();

  // ---- per-row mode over 128 indices (one wave32 per row, 2 rows/wave) --
  const int wave = t >> 5;
  const int lane = t & 31;
#pragma unroll
  for (int rr = 0; rr < 2; ++rr) {
    const int row = wave * 2 + rr;
    for (int which = 0; which < 2; ++which) {
      const int* a = which ? (sIMax + row * DIM) : (sIMin + row * DIM);
      int myv[4], cnt[4];
#pragma unroll
      for (int q = 0; q < 4; ++q) { myv[q] = a[lane * 4 + q]; cnt[q] = 0; }
      for (int j = 0; j < DIM; ++j) {
        const int v = a[j];                        // wave-uniform LDS broadcast
#pragma unroll
        for (int q = 0; q < 4; ++q) cnt[q] += (v == myv[q]) ? 1 : 0;
      }
      // key: maximize count, tie-break smallest index (idx < 1024 -> 10 bits)
      int key = 0;
#pragma unroll
      for (int q = 0; q < 4; ++q) {
        const int kq = (cnt[q] << 10) | (1023 - myv[q]);
        key = key > kq ? key : kq;
      }
#pragma unroll
      for (int off = 16; off > 0; off >>= 1) {
        const int o = __shfl_xor(key, off, 32);    // wave32 shuffle
        key = key > o ? key : o;
      }
      const int mode = 1023 - (key & 1023);
      if (lane == 0) { if (which) sModeN[row] = mode; else sModeP[row] = mode; }
    }
  }
  __syncthreads();

  // ---- triplet margin loss (swap=True) per row --------------------------
#pragma unroll
  for (int rr = 0; rr < 2; ++rr) {
    const int row = wave * 2 + rr;
    const int pi = sModeP[row];
    const int ni = sModeN[row];
    const float4 xq = *(const float4*)(&sX[row * DIM + lane * 4]);
    const float4 pq = *(const float4*)(cent + (size_t)pi * DIM + lane * 4);
    const float4 nq = *(const float4*)(cent + (size_t)ni * DIM + lane * 4);
    const float xa[4] = {xq.x, xq.y, xq.z, xq.w};
    const float pa[4] = {pq.x, pq.y, pq.z, pq.w};
    const float na[4] = {nq.x, nq.y, nq.z, nq.w};
    float sp = 0.f, sn = 0.f, ss = 0.f;
#pragma unroll
    for (int q = 0; q < 4; ++q) {
      const float ap = xa[q] - pa[q] + EPS; sp += ap * ap;
      const float an = xa[q] - na[q] + EPS; sn += an * an;
      const float aw = pa[q] - na[q] + EPS; ss += aw * aw;
    }
#pragma unroll
    for (int off = 16; off > 0; off >>= 1) {       // fixed-order, deterministic
      sp += __shfl_xor(sp, off, 32);
      sn += __shfl_xor(sn, off, 32);
      ss += __shfl_xor(ss, off, 32);
    }
    if (lane == 0) {
      const float dp = sqrtf(sp);
      float       dn = sqrtf(sn);
      const float dw = sqrtf(ss);
      dn = fminf(dn, dw);
      const float l = dp - dn + MARGIN;
      sLoss[row] = l > 0.f ? l : 0.f;
    }
  }
  __syncthreads();

  if (t == 0) {
    float acc = 0.f;
    for (int r = 0; r < ROWS; ++r) acc += sLoss[r];   // fixed order
    block_loss[blockIdx.x] = acc;
  }
}

__global__ __launch_bounds__(256)
void ctl_reduce_kernel(const float* __restrict__ block_loss,
                       float* __restrict__ out) {
  __shared__ float s[256];
  const int t = threadIdx.x;
  s[t] = block_loss[t];                 // exactly 256 partials
  __syncthreads();
  for (int off = 128; off > 0; off >>= 1) {
    if (t < off) s[t] += s[t + off];    // fixed-order tree -> deterministic
    __syncthreads();
  }
  if (t == 0) out[0] = s[0] * (1.0f / (float)B_ROWS);
}

extern "C" void kernel_launch(void* const* d_in, const int* in_sizes, int n_in,
                              void* d_out, int out_size, void* d_ws, size_t ws_size,
                              hipStream_t stream) {
  (void)in_sizes; (void)n_in; (void)out_size; (void)ws_size;
  const float* x    = (const float*)d_in[0];   // [4096, 128] f32
  const float* cent = (const float*)d_in[1];   // [1024, 128] f32
  float*       out  = (float*)d_out;           // scalar f32
  float*       ws   = (float*)d_ws;            // 256 block partials (1 KB)

  ctl_main_kernel<<<B_ROWS / ROWS, 256, 0, stream>>>(x, cent, ws);
  ctl_reduce_kernel<<<1, 256, 0, stream>>>(ws, out);
}